// GCN_58901181497836
// MI455X (gfx1250) — compile-verified
//
#include <hip/hip_runtime.h>
#include <hip/hip_bf16.h>
#include <math.h>

typedef float v2f __attribute__((ext_vector_type(2)));
typedef float v8f __attribute__((ext_vector_type(8)));

#define N_NODES 100000
#define N_EDGES 1600000
#define CH      256
#define OUTC    40

// ---------------------------------------------------------------- degree ----
__global__ void fill_ones_k(float* deg, int n) {
    int i = blockIdx.x * blockDim.x + threadIdx.x;
    if (i < n) deg[i] = 1.0f;                      // self-loop contributes +1
}

__global__ void deg_count_k(const int* __restrict__ dst, float* deg, int nE) {
    int i = blockIdx.x * blockDim.x + threadIdx.x;
    if (i < nE) atomicAdd(&deg[dst[i]], 1.0f);
}

__global__ void deg_to_dinv_k(float* deg, int n) {
    int i = blockIdx.x * blockDim.x + threadIdx.x;
    if (i < n) deg[i] = rsqrtf(deg[i]);
}

// ------------------------------------------------------- WMMA fp32 GEMM -----
// H[n_rows x n_cols] = X[n_rows x 256] * W[256 x n_cols]
// block = 256 threads (8 waves). Block tile 64x32, wave tile 16x16.
// Uses V_WMMA_F32_16X16X4_F32, K staged through LDS in chunks of 64.
#define TM 64
#define TN 32
#define TK 64

__global__ __launch_bounds__(256)
void gemm_wmma_f32(const float* __restrict__ X, const float* __restrict__ W,
                   float* __restrict__ H, int n_rows, int n_cols) {
    __shared__ float xs[TM][TK + 4];   // +4 pad: conflict-free A-frag reads
    __shared__ float ws[TK][TN + 4];

    const int tid  = threadIdx.x;
    const int lane = tid & 31;
    const int wv   = tid >> 5;
    const int khi  = lane >> 4;        // 0: K pair {0,1}; 1: K pair {2,3}
    const int l15  = lane & 15;
    const int m_blk = blockIdx.x * TM;
    const int n_blk = blockIdx.y * TN;
    const int m_sub = (wv & 3) * 16;
    const int n_sub = (wv >> 2) * 16;

    v8f acc = {};

    for (int kc = 0; kc < 256; kc += TK) {
        // --- stage X tile (64x64 f32): 4 x float4 per thread, coalesced ---
        #pragma unroll
        for (int i = 0; i < 4; ++i) {
            int flat = tid * 4 + i * 1024;
            int r = flat >> 6;
            int c = flat & 63;
            int gr = m_blk + r;
            float4 v = make_float4(0.f, 0.f, 0.f, 0.f);
            if (gr < n_rows)
                v = *(const float4*)(X + (size_t)gr * 256 + kc + c);
            xs[r][c + 0] = v.x; xs[r][c + 1] = v.y;
            xs[r][c + 2] = v.z; xs[r][c + 3] = v.w;
        }
        // --- stage W tile (64x32 f32): 8 scalars/thread, col-guarded ------
        #pragma unroll
        for (int i = 0; i < 8; ++i) {
            int flat = tid + i * 256;
            int r = flat >> 5;
            int c = flat & 31;
            int gc = n_blk + c;
            float v = 0.f;
            if (gc < n_cols)
                v = W[(size_t)(kc + r) * n_cols + gc];
            ws[r][c] = v;
        }
        __syncthreads();

        // --- 16 WMMA steps over this K chunk ------------------------------
        #pragma unroll
        for (int t = 0; t < 16; ++t) {
            const int ks = 4 * t;
            // A 16x4 layout: lanes 0-15 rows M, VGPR0/1 = K {0,1}; lanes 16-31 = K {2,3}
            v2f a;
            a.x = xs[m_sub + l15][ks + 2 * khi + 0];
            a.y = xs[m_sub + l15][ks + 2 * khi + 1];
            // B 4x16 layout: row striped across lanes per VGPR, upper lane half = K+2
            v2f b;
            b.x = ws[ks + 2 * khi + 0][n_sub + l15];
            b.y = ws[ks + 2 * khi + 1][n_sub + l15];
            acc = __builtin_amdgcn_wmma_f32_16x16x4_f32(
                      false, a, false, b, (short)0, acc, false, false);
        }
        __syncthreads();
    }

    // --- store D: VGPR r -> row m_sub + r + 8*khi, col n_sub + l15 --------
    const int col = n_blk + n_sub + l15;
    if (col < n_cols) {
        #pragma unroll
        for (int r = 0; r < 8; ++r) {
            int row = m_blk + m_sub + r + 8 * khi;
            if (row < n_rows)
                H[(size_t)row * n_cols + col] = acc[r];
        }
    }
}

// ------------------------------------------- self-loop init of aggregate ----
template <int C>
__global__ void self_init_k(const float* __restrict__ h,
                            const float* __restrict__ dinv,
                            float* __restrict__ agg, int n) {
    long i = (long)blockIdx.x * blockDim.x + threadIdx.x;
    if (i < (long)n * C) {
        int node = (int)(i / C);
        float di = dinv[node];
        agg[i] = h[i] * di * di;
    }
}

// ------------------------------------------------- edge gather/scatter ------
// one wave per edge: coalesced 128B strips of h[src], L2-resident fp32 atomics
template <int C>
__global__ void edge_agg_k(const float* __restrict__ h,
                           const int* __restrict__ src,
                           const int* __restrict__ dst,
                           const float* __restrict__ dinv,
                           float* __restrict__ agg, int nE) {
    const int lane   = threadIdx.x & 31;
    const int wid    = (int)((blockIdx.x * (long)blockDim.x + threadIdx.x) >> 5);
    const int nWaves = (int)((gridDim.x * (long)blockDim.x) >> 5);
    for (int e = wid; e < nE; e += nWaves) {
        const int s = src[e];
        const int d = dst[e];
        const float norm = dinv[s] * dinv[d];
        #pragma unroll
        for (int c0 = 0; c0 < C; c0 += 32) {
            int c = c0 + lane;
            if ((C & 31) == 0 || c < C)
                atomicAdd(&agg[(size_t)d * C + c], h[(size_t)s * C + c] * norm);
        }
    }
}

// ------------------------------------------------------ bias + relu ---------
__global__ void finish_relu_k(const float* __restrict__ agg,
                              const float* __restrict__ bias,
                              float* __restrict__ out, int n) {
    long i = (long)blockIdx.x * blockDim.x + threadIdx.x;
    if (i < (long)n * CH) {
        float v = agg[i] + bias[i & (CH - 1)];
        out[i] = v > 0.f ? v : 0.f;
    }
}

// ------------------------------------------ bias + log_softmax (C=40) -------
__global__ void final_logsoftmax_k(const float* __restrict__ agg,
                                   const float* __restrict__ b3,
                                   float* __restrict__ out, int n) {
    const int lane = threadIdx.x & 31;
    const int node = (int)((blockIdx.x * (long)blockDim.x + threadIdx.x) >> 5);
    if (node >= n) return;
    const float* a = agg + (size_t)node * OUTC;
    float x0 = a[lane] + b3[lane];                                 // c = 0..31
    float x1 = (lane < OUTC - 32) ? (a[32 + lane] + b3[32 + lane]) // c = 32..39
                                  : -INFINITY;
    float m = fmaxf(x0, x1);
    #pragma unroll
    for (int off = 16; off > 0; off >>= 1) m = fmaxf(m, __shfl_xor(m, off, 32));
    float s = expf(x0 - m) + ((lane < OUTC - 32) ? expf(x1 - m) : 0.f);
    #pragma unroll
    for (int off = 16; off > 0; off >>= 1) s += __shfl_xor(s, off, 32);
    float lse = m + logf(s);
    out[(size_t)node * OUTC + lane] = x0 - lse;
    if (lane < OUTC - 32)
        out[(size_t)node * OUTC + 32 + lane] = x1 - lse;
}

// ---------------------------------------------------------------------------
extern "C" void kernel_launch(void* const* d_in, const int* in_sizes, int n_in,
                              void* d_out, int out_size, void* d_ws, size_t ws_size,
                              hipStream_t stream) {
    const float* x   = (const float*)d_in[0];
    const int*   ei  = (const int*)d_in[1];
    const int*   src = ei;
    const int*   dst = ei + N_EDGES;
    const float* W1  = (const float*)d_in[2];
    const float* b1  = (const float*)d_in[3];
    const float* W2  = (const float*)d_in[4];
    const float* b2  = (const float*)d_in[5];
    const float* W3  = (const float*)d_in[6];
    const float* b3  = (const float*)d_in[7];
    float* out = (float*)d_out;

    float* bufA = (float*)d_ws;                        // N x 256
    float* bufB = bufA + (size_t)N_NODES * CH;         // N x 256
    float* dinv = bufB + (size_t)N_NODES * CH;         // N

    const int B = 256;
    const int gN  = (N_NODES + B - 1) / B;
    const int gE  = (N_EDGES + B - 1) / B;
    const int gNC = ((long)N_NODES * CH + B - 1) / B;
    const dim3 gemm_g256((N_NODES + TM - 1) / TM, CH / TN);
    const dim3 gemm_g40((N_NODES + TM - 1) / TM, (OUTC + TN - 1) / TN);
    const int edge_blocks = 5120;                      // 40960 waves

    // degrees -> dinv
    fill_ones_k<<<gN, B, 0, stream>>>(dinv, N_NODES);
    deg_count_k<<<gE, B, 0, stream>>>(dst, dinv, N_EDGES);
    deg_to_dinv_k<<<gN, B, 0, stream>>>(dinv, N_NODES);

    // ---- layer 1: h1 = x@W1 (A), agg (B), relu -> A ----
    gemm_wmma_f32<<<gemm_g256, B, 0, stream>>>(x, W1, bufA, N_NODES, CH);
    self_init_k<CH><<<gNC, B, 0, stream>>>(bufA, dinv, bufB, N_NODES);
    edge_agg_k<CH><<<edge_blocks, B, 0, stream>>>(bufA, src, dst, dinv, bufB, N_EDGES);
    finish_relu_k<<<gNC, B, 0, stream>>>(bufB, b1, bufA, N_NODES);

    // ---- layer 2: h2 = A@W2 (B), agg (A), relu -> B ----
    gemm_wmma_f32<<<gemm_g256, B, 0, stream>>>(bufA, W2, bufB, N_NODES, CH);
    self_init_k<CH><<<gNC, B, 0, stream>>>(bufB, dinv, bufA, N_NODES);
    edge_agg_k<CH><<<edge_blocks, B, 0, stream>>>(bufB, src, dst, dinv, bufA, N_EDGES);
    finish_relu_k<<<gNC, B, 0, stream>>>(bufA, b2, bufB, N_NODES);

    // ---- layer 3: h3 = B@W3 -> A[0..N*40), agg3 -> A + N*40 ----
    float* h3   = bufA;
    float* agg3 = bufA + (size_t)N_NODES * OUTC;
    const int gN40 = ((long)N_NODES * OUTC + B - 1) / B;
    gemm_wmma_f32<<<gemm_g40, B, 0, stream>>>(bufB, W3, h3, N_NODES, OUTC);
    self_init_k<OUTC><<<gN40, B, 0, stream>>>(h3, dinv, agg3, N_NODES);
    edge_agg_k<OUTC><<<edge_blocks, B, 0, stream>>>(h3, src, dst, dinv, agg3, N_EDGES);

    // ---- bias + log_softmax, one wave per node ----
    final_logsoftmax_k<<<(N_NODES * 32 + B - 1) / B, B, 0, stream>>>(agg3, b3, out, N_NODES);
}